// VibrationFeature_84121229459701
// MI455X (gfx1250) — compile-verified
//
#include <hip/hip_runtime.h>
#include <math.h>
#include <stdint.h>

#define BATCH   8192
#define UNITS   16384
#define TPB     256
#define CHUNK_F 1024                    // floats per chunk (4 KB)
#define NCHUNK  (UNITS / CHUNK_F)       // 16
#define DEPTH   4                       // async pipeline depth (LDS ring slots)

// ---- CDNA5 async global->LDS copy (ASYNCcnt-tracked), inline asm ----
__device__ __forceinline__ void async_ld_b128(uint32_t lds_byte, const void* gptr) {
    asm volatile("global_load_async_to_lds_b128 %0, %1, off"
                 :: "v"(lds_byte), "v"((uint64_t)(uintptr_t)gptr)
                 : "memory");
}
__device__ __forceinline__ void wait_async_le3() {
    asm volatile("s_wait_asynccnt 0x3" ::: "memory");
}
__device__ __forceinline__ void wait_async_0() {
    asm volatile("s_wait_asynccnt 0x0" ::: "memory");
}
__device__ __forceinline__ void wait_ds_0() {
    asm volatile("s_wait_dscnt 0x0" ::: "memory");
}

__global__ __launch_bounds__(TPB)
void vibration_features_kernel(const float* __restrict__ x,
                               const float* __restrict__ w,
                               const float* __restrict__ bias,
                               float* __restrict__ out) {
    __shared__ float4 lbuf[DEPTH * TPB];   // 16 KB ring buffer
    __shared__ float  red[8 * 8];          // 8 waves x 8 partials

    const int tid = threadIdx.x;
    const int row = blockIdx.x;

    const float4* xrow = (const float4*)(x + (size_t)row * UNITS);
    const float4* w4   = (const float4*)w;

    // LDS byte address of this thread's slot in ring buffer 0.
    // Generic LDS pointers are {SHARED_BASE_hi32, lds_offset}; low 32 bits are
    // the per-workgroup LDS byte offset the async engine expects.
    const uint32_t lds_base = (uint32_t)(uintptr_t)(&lbuf[tid]);
    const uint32_t buf_stride = TPB * (uint32_t)sizeof(float4);

    // ---- prologue: fill the pipeline ----
#pragma unroll
    for (int k = 0; k < DEPTH; ++k)
        async_ld_b128(lds_base + (uint32_t)k * buf_stride,
                      xrow + k * (CHUNK_F / 4) + tid);

    float s1 = 0.f, s2 = 0.f, s3 = 0.f, s4 = 0.f, sa = 0.f, sr = 0.f;
    float mx = -__builtin_inff(), mn = __builtin_inff();

    auto process = [&](int k) {
        float4 v  = lbuf[(k & (DEPTH - 1)) * TPB + tid];   // ds_load_b128 (own slot)
        float4 wv = w4[k * (CHUNK_F / 4) + tid];           // L2-resident, tiny
        float t[4] = { v.x * wv.x, v.y * wv.y, v.z * wv.z, v.w * wv.w };
#pragma unroll
        for (int i = 0; i < 4; ++i) {
            float u  = t[i];
            mx = fmaxf(mx, u);
            mn = fminf(mn, u);
            s1 += u;
            float u2 = u * u;
            s2 = fmaf(u, u, s2);
            s3 = fmaf(u2, u, s3);
            s4 = fmaf(u2, u2, s4);
            float a = fabsf(u);
            sa += a;
            sr += sqrtf(a);
        }
    };

    // ---- steady state: wait depth-1, process, refill the slot just drained ----
#pragma unroll
    for (int k = 0; k < NCHUNK - DEPTH; ++k) {
        wait_async_le3();                 // chunk k landed in LDS
        process(k);
        wait_ds_0();                      // WAR: our ds_loads done before overwrite
        async_ld_b128(lds_base + (uint32_t)(k & (DEPTH - 1)) * buf_stride,
                      xrow + (k + DEPTH) * (CHUNK_F / 4) + tid);
    }
    // ---- drain ----
    wait_async_0();
#pragma unroll
    for (int k = NCHUNK - DEPTH; k < NCHUNK; ++k)
        process(k);

    // ---- wave32 cross-lane reduction ----
#pragma unroll
    for (int off = 16; off >= 1; off >>= 1) {
        s1 += __shfl_xor(s1, off);
        s2 += __shfl_xor(s2, off);
        s3 += __shfl_xor(s3, off);
        s4 += __shfl_xor(s4, off);
        sa += __shfl_xor(sa, off);
        sr += __shfl_xor(sr, off);
        mx = fmaxf(mx, __shfl_xor(mx, off));
        mn = fminf(mn, __shfl_xor(mn, off));
    }
    const int wave = tid >> 5;
    if ((tid & 31) == 0) {
        red[wave * 8 + 0] = s1; red[wave * 8 + 1] = s2;
        red[wave * 8 + 2] = s3; red[wave * 8 + 3] = s4;
        red[wave * 8 + 4] = sa; red[wave * 8 + 5] = sr;
        red[wave * 8 + 6] = mx; red[wave * 8 + 7] = mn;
    }
    __syncthreads();

    if (tid == 0) {
        float S1 = 0.f, S2 = 0.f, S3 = 0.f, S4 = 0.f, SA = 0.f, SR = 0.f;
        float MX = -__builtin_inff(), MN = __builtin_inff();
#pragma unroll
        for (int i = 0; i < 8; ++i) {
            S1 += red[i * 8 + 0]; S2 += red[i * 8 + 1];
            S3 += red[i * 8 + 2]; S4 += red[i * 8 + 3];
            SA += red[i * 8 + 4]; SR += red[i * 8 + 5];
            MX = fmaxf(MX, red[i * 8 + 6]);
            MN = fminf(MN, red[i * 8 + 7]);
        }
        const float inv_n = 1.0f / (float)UNITS;
        const float mu  = S1 * inv_n;          // E[v]
        const float e2  = S2 * inv_n;          // E[v^2]
        const float e3  = S3 * inv_n;          // E[v^3]
        const float e4  = S4 * inv_n;          // E[v^4]
        const float rms = sqrtf(e2);

        const float ptp       = fabsf(MX - MN);
        const float crest     = MX / rms;
        const float shape     = rms / (SA * inv_n);
        const float impulse   = MX / e2;               // faithful: max / mean-square
        const float clearance = MX / (SR * inv_n);

        const float mu2 = mu * mu;
        const float m2  = e2 - mu2;
        const float m3  = e3 - 3.f * mu * e2 + 2.f * mu * mu2;
        const float m4  = e4 - 4.f * mu * e3 + 6.f * mu2 * e2 - 3.f * mu2 * mu2;

        float thr = 1e-6f * mu;                 // finfo(float32).resolution
        thr *= thr;
        const bool  zero = (m2 <= thr);
        const float nanv = __int_as_float(0x7fc00000);
        const float kurt = (zero ? nanv : m4 / (m2 * m2)) - 3.0f;
        const float skew =  zero ? nanv : m3 / (m2 * sqrtf(m2));

        float* o = out + (size_t)row * 7;
        o[0] = ptp       * bias[0];
        o[1] = crest     * bias[1];
        o[2] = shape     * bias[2];
        o[3] = impulse   * bias[3];
        o[4] = clearance * bias[4];
        o[5] = kurt      * bias[5];
        o[6] = skew      * bias[6];
    }
}

extern "C" void kernel_launch(void* const* d_in, const int* in_sizes, int n_in,
                              void* d_out, int out_size, void* d_ws, size_t ws_size,
                              hipStream_t stream) {
    (void)in_sizes; (void)n_in; (void)d_ws; (void)ws_size; (void)out_size;
    const float* x    = (const float*)d_in[0];
    const float* w    = (const float*)d_in[1];
    const float* bias = (const float*)d_in[2];
    float*       out  = (float*)d_out;
    vibration_features_kernel<<<BATCH, TPB, 0, stream>>>(x, w, bias, out);
}